// MiniBatchDiscrimination_68487548502486
// MI455X (gfx1250) — compile-verified
//
#include <hip/hip_runtime.h>
#include <hip/hip_bf16.h>

// ---------------------------------------------------------------------------
// MiniBatchDiscrimination on MI455X (gfx1250, wave32)
//   Tt = pad+transpose(T)          -> one-time 2MB LDS-tiled transpose
//   m  = x @ T                     -> WMMA f32 16x16x4, branch-free K-loop,
//                                     both operands as global_load_b64
//   feat[i,k] = sum_j exp(-L1)     -> VALU + v_exp_f32, LDS broadcast reads
//   out = [x | feat]
// ---------------------------------------------------------------------------

typedef __attribute__((ext_vector_type(2))) float v2f;
typedef __attribute__((ext_vector_type(8))) float v8f;

#define BATCH   1024
#define FDIM    1024
#define NKER    100
#define DPK     5
#define NCOLS   (NKER * DPK)     // 500
#define NPAD    512              // padded N (columns of m / rows of Tt)
#define OUTW    (FDIM + NKER)    // 1124

// ---------------------------------------------------------------------------
// Kernel 0: Tt[512][1024] = transpose(T[1024][500]) with zero-padded rows
// 32x32 tiles via LDS (33-wide to kill bank conflicts); both global sides
// coalesced. One-shot 2 MB pre-pass — negligible vs. the pairwise stage.
// ---------------------------------------------------------------------------
__global__ __launch_bounds__(256) void pad_transpose_T(
    const float* __restrict__ T, float* __restrict__ Tt)
{
    __shared__ float tile[32][33];
    const int tx = threadIdx.x & 31;          // 0..31
    const int ty = threadIdx.x >> 5;          // 0..7
    const int kb = blockIdx.x * 32;           // K base (32 blocks over 1024)
    const int nb = blockIdx.y * 32;           // N base (16 blocks over 512)

#pragma unroll
    for (int r = 0; r < 32; r += 8) {
        const int k = kb + ty + r;
        const int n = nb + tx;
        tile[ty + r][tx] = (n < NCOLS) ? T[k * NCOLS + n] : 0.0f;
    }
    __syncthreads();
#pragma unroll
    for (int r = 0; r < 32; r += 8) {
        const int n = nb + ty + r;            // Tt row  = n
        const int k = kb + tx;                // Tt col  = k
        Tt[n * FDIM + k] = tile[tx][ty + r];
    }
}

// ---------------------------------------------------------------------------
// Kernel 1: m[1024][512] = x[1024][1024] @ Tt^T   (WMMA f32 16x16x4)
// One wave32 per 16x16 tile; K-loop of 256 steps, each: 2x global_load_b64
// + 1x v_wmma_f32_16x16x4_f32, fully branch-free (EXEC stays all-1s).
//
// A (16x4 f32, 2 VGPRs): lane L: M = L&15; VGPR0/1 = K+(L<16?0:2), +1.
// B (4x16 f32, 2 VGPRs): lane L: N = L&15; VGPR0/1 = K+(L<16?0:2), +1.
//   (Tt is N-major, so B's per-lane pair is contiguous -> b64 load.)
// C/D (16x16 f32, 8 VGPRs): VGPR r: lanes 0-15 -> M=r, lanes 16-31 -> M=r+8.
// ---------------------------------------------------------------------------
__global__ __launch_bounds__(256) void gemm_wmma_f32(
    const float* __restrict__ x, const float* __restrict__ Tt,
    float* __restrict__ m)
{
    const int lane    = threadIdx.x & 31;
    const int wave    = threadIdx.x >> 5;
    const int wid     = blockIdx.x * 8 + wave;   // 0..2047
    const int tileM   = wid >> 5;                // 0..63
    const int tileN   = wid & 31;                // 0..31
    const int halfSel = lane >> 4;               // 0 | 1 (half-wave K pair)
    const int lrow    = lane & 15;

    const float* aPtr = x  + (tileM * 16 + lrow) * FDIM + halfSel * 2;
    const float* bPtr = Tt + (tileN * 16 + lrow) * FDIM + halfSel * 2;

    v8f acc = {};
#pragma unroll 8
    for (int kb = 0; kb < FDIM; kb += 4) {
        const v2f a = *(const v2f*)(aPtr + kb);   // global_load_b64
        const v2f b = *(const v2f*)(bPtr + kb);   // global_load_b64
        // (neg_a, A, neg_b, B, c_mod, C, reuse_a, reuse_b)
        acc = __builtin_amdgcn_wmma_f32_16x16x4_f32(
            false, a, false, b, (short)0, acc, false, false);
    }

    // Store D-tile (padded width NPAD -> no column guard).
    float* dPtr = m + (tileM * 16 + halfSel * 8) * NPAD + tileN * 16 + lrow;
#pragma unroll
    for (int r = 0; r < 8; ++r)
        dPtr[r * NPAD] = acc[r];
}

// ---------------------------------------------------------------------------
// Kernel 2: feat[i,k] = sum_j exp(-sum_d |m[i,k,d] - m[j,k,d]|)
// Grid: (k = 0..99, ichunk = 0..3), block = 256 threads (8 waves).
// All 1024 j-rows of kernel k staged in LDS (stride-8 pad -> 32 KB, b128-
// aligned). The j loop is wave-uniform -> LDS broadcast reads, no conflicts.
// ---------------------------------------------------------------------------
__global__ __launch_bounds__(256) void pairwise_expl1(
    const float* __restrict__ m, float* __restrict__ out)
{
    __shared__ float sm[BATCH * 8];              // 32 KB, [j][8] padded

    const int k = blockIdx.x;                    // kernel index 0..99
    const int i = blockIdx.y * 256 + threadIdx.x;
    const int kbase = k * DPK;

    // Stage m[:, k*5 : k*5+5] into LDS (scalar loads: k*5 not 16B aligned).
    for (int j = threadIdx.x; j < BATCH; j += 256) {
        const float* src = m + j * NPAD + kbase;
#pragma unroll
        for (int d = 0; d < DPK; ++d)
            sm[j * 8 + d] = src[d];
    }
    __syncthreads();

    // This thread's own 5 projections (L2-resident workspace).
    const float* mi = m + i * NPAD + kbase;
    const float m0 = mi[0], m1 = mi[1], m2 = mi[2], m3 = mi[3], m4 = mi[4];

    float acc = 0.0f;
#pragma unroll 4
    for (int j = 0; j < BATCH; ++j) {
        const float4 v = *(const float4*)&sm[j * 8];   // ds_load_b128 (bcast)
        const float  v4 = sm[j * 8 + 4];               // ds_load_b32  (bcast)
        const float l1 = __builtin_fabsf(m0 - v.x)
                       + __builtin_fabsf(m1 - v.y)
                       + __builtin_fabsf(m2 - v.z)
                       + __builtin_fabsf(m3 - v.w)
                       + __builtin_fabsf(m4 - v4);
        acc += __expf(-l1);                            // v_exp_f32 (trans pipe)
    }

    out[i * OUTW + FDIM + k] = acc;
}

// ---------------------------------------------------------------------------
// Kernel 3: out[:, 0:1024] = x   (float4 copy; OUTW = 1124 = 4*281 -> aligned)
// ---------------------------------------------------------------------------
__global__ __launch_bounds__(256) void copy_x_rows(
    const float* __restrict__ x, float* __restrict__ out)
{
    const int row = blockIdx.x;
    const float4* src = (const float4*)(x + row * FDIM);
    float4*       dst = (float4*)(out + row * OUTW);
    dst[threadIdx.x] = src[threadIdx.x];             // 256 threads * float4
}

// ---------------------------------------------------------------------------
extern "C" void kernel_launch(void* const* d_in, const int* in_sizes, int n_in,
                              void* d_out, int out_size, void* d_ws, size_t ws_size,
                              hipStream_t stream)
{
    const float* x = (const float*)d_in[0];   // [1024, 1024] f32
    const float* T = (const float*)d_in[1];   // [1024, 500]  f32
    float* out = (float*)d_out;               // [1024, 1124] f32

    float* ws = (float*)d_ws;
    float* m  = ws;                           // [1024][512] f32  (2 MB)
    float* Tt = ws + BATCH * NPAD;            // [512][1024] f32  (2 MB)

    // 0) Pad + transpose T so the GEMM K-loop is guard-free and b64-loadable.
    pad_transpose_T<<<dim3(FDIM / 32, NPAD / 32), dim3(256), 0, stream>>>(T, Tt);

    // 1) GEMM: 64x32 tiles of 16x16, 8 waves per 256-thread block.
    gemm_wmma_f32<<<dim3(256), dim3(256), 0, stream>>>(x, Tt, m);

    // 2) Pairwise exp(-L1) reduction: 100 kernels x 4 row-chunks.
    pairwise_expl1<<<dim3(NKER, 4), dim3(256), 0, stream>>>(m, out);

    // 3) Copy x into the first 1024 output columns.
    copy_x_rows<<<dim3(BATCH), dim3(256), 0, stream>>>(x, out);
}